// MambaNextToken_3599182594852
// MI455X (gfx1250) — compile-verified
//
#include <hip/hip_runtime.h>
#include <math.h>

// ---------------------------------------------------------------------------
// Mamba block forward for MI455X (gfx1250, wave32, WMMA, async->LDS).
// B=4, D_MODEL=1024, L=2048, D_INNER=2048, D_STATE=16, D_CONV=4, DT_RANK=64.
//  - GEMMs on the matrix pipe: V_WMMA_F32_16X16X4_F32 (fp32-exact vs ref).
//  - Selective scan: chunked linear-recurrence scan (32 chunks x 64 steps):
//      pass1 per-chunk local scan + propagator, pass2 32-step combine,
//      pass3 re-scan with true initial state + fused epilogue.
//    B/C slabs staged via GLOBAL_LOAD_ASYNC_TO_LDS_B128 (ASYNCcnt path).
// ---------------------------------------------------------------------------

typedef __attribute__((ext_vector_type(2))) float v2f;
typedef __attribute__((ext_vector_type(8))) float v8f;

#define MB 4
#define DMODEL 1024
#define LSEQ 2048
#define DINNER 2048
#define DSTATE 16
#define DTRANK 64
#define XPN (DTRANK + 2 * DSTATE) // 96
#define MROWS (MB * LSEQ)         // 8192
#define NCHUNK 32
#define CLEN (LSEQ / NCHUNK)      // 64

__device__ __forceinline__ float silu_f(float v) {
    return v / (1.0f + expf(-v));
}

// Async copy of 16 contiguous bytes global -> LDS (wave32, ASYNCcnt-tracked).
__device__ __forceinline__ void async_g2l_b128(const float* g, float* lds) {
    const unsigned lds_off = (unsigned)(size_t)(void*)lds; // LDS offset: low 32 bits of the pointer
    const unsigned long long ga = (unsigned long long)(size_t)g;
    asm volatile("global_load_async_to_lds_b128 %0, %1, off"
                 :
                 : "v"(lds_off), "v"(ga)
                 : "memory");
}
__device__ __forceinline__ void async_wait0() {
    asm volatile("s_wait_asynccnt 0x0" ::: "memory");
}

// ---------------------------------------------------------------------------
// WMMA fp32 GEMM:  C[M,N] = A[M,K] * W[N,K]^T  (+bias, +softplus, opt. T-store)
// AMODE 0: A[m*lda + k] row-major.
// AMODE 1: A is x (B, D_MODEL, L); row m = b*L + l, elem (m,k) at
//          x[(b*D_MODEL + k)*L + l]  (transpose folded into loads).
// EPI 0: none; 1: +bias then softplus; 2: +bias.
// SM  0: C[m*N + n]; 1: transposed store C[(b*N + n)*L + l]  (m = b*L + l).
// Block = 256 threads (8 waves); each wave owns a 16(M) x 64(N) tile.
// ---------------------------------------------------------------------------
template <int AMODE, int EPI, int SM>
__global__ void __launch_bounds__(256)
wmma_gemm_f32(const float* __restrict__ A, const float* __restrict__ W,
              const float* __restrict__ bias, float* __restrict__ C,
              int M, int N, int K, int lda) {
    const int lane = threadIdx.x & 31;
    const int wave = threadIdx.x >> 5;
    const int m0   = blockIdx.y * 128 + wave * 16;
    const int n0   = blockIdx.x * 64;
    const int half = lane >> 4;
    const int row  = lane & 15;

    // Clamp B row indices so EXEC stays all-ones (WMMA requirement).
    int ncl[4];
#pragma unroll
    for (int t = 0; t < 4; ++t) {
        int n = n0 + t * 16 + row;
        ncl[t] = (n < N) ? n : (N - 1);
    }

    v8f acc[4] = {};

    for (int k = 0; k < K; k += 4) {
        if ((k & 127) == 0 && (k + 256) < K) {
            __builtin_prefetch(W + (size_t)ncl[0] * K + k + 256, 0, 1);
        }
        // A tile 16x4: VGPR0={K=k (l0-15), K=k+2 (l16-31)}, VGPR1={k+1, k+3}
        v2f a;
        if (AMODE == 0) {
            a = *(const v2f*)(A + (size_t)(m0 + row) * lda + k + 2 * half);
        } else {
            const int m = m0 + row;
            const int b = m >> 11;
            const int l = m & (LSEQ - 1);
            const float* px =
                A + ((size_t)(b * DMODEL + k + 2 * half) << 11) + l;
            a.x = px[0];
            a.y = px[(size_t)1 << 11];
        }
        // B tiles 4x16: N across lanes, K across lane-half + VGPR pair
#pragma unroll
        for (int t = 0; t < 4; ++t) {
            v2f bv = *(const v2f*)(W + (size_t)ncl[t] * K + k + 2 * half);
            acc[t] = __builtin_amdgcn_wmma_f32_16x16x4_f32(
                false, a, false, bv, (short)0, acc[t], false, false);
        }
    }

    // C/D layout: VGPR i holds M=i (lanes0-15), M=i+8 (lanes16-31)
#pragma unroll
    for (int t = 0; t < 4; ++t) {
        const int col = n0 + t * 16 + row;
        if (col < N) {
#pragma unroll
            for (int i = 0; i < 8; ++i) {
                const int r = m0 + i + 8 * half;
                float v = acc[t][i];
                if (EPI != 0) v += bias[col];
                if (EPI == 1) v = (v > 20.0f) ? v : log1pf(expf(v));
                if (SM == 0) {
                    C[(size_t)r * N + col] = v;
                } else {
                    const int b = r >> 11;
                    const int l = r & (LSEQ - 1);
                    C[((size_t)(b * N + col) << 11) + l] = v;
                }
            }
        }
    }
}

// ---------------------------------------------------------------------------
// Depthwise causal conv1d (D_CONV=4) + SiLU.  u = xz[:, 0:DINNER].
// ---------------------------------------------------------------------------
__global__ void __launch_bounds__(256)
conv_silu_kernel(const float* __restrict__ xz, const float* __restrict__ cw,
                 const float* __restrict__ cb, float* __restrict__ ucv) {
    const size_t idx = (size_t)blockIdx.x * blockDim.x + threadIdx.x;
    if (idx >= (size_t)MROWS * DINNER) return;
    const int e     = (int)(idx & (DINNER - 1));
    const size_t ml = idx >> 11;
    const int l     = (int)(ml & (LSEQ - 1));
    float acc = cb[e];
#pragma unroll
    for (int k = 0; k < 4; ++k) {
        const int ls = l - 3 + k;
        if (ls >= 0)
            acc += cw[e * 4 + k] * xz[(ml - (size_t)(3 - k)) * (2 * DINNER) + e];
    }
    ucv[ml * DINNER + e] = silu_f(acc);
}

// ---------------------------------------------------------------------------
// Chunk-slab preload: xdbl[m0..m0+CLEN-1][64..95] -> sBC[CLEN*32] via async
// b128 transfers (16B each, 16B-aligned: row stride 384B, col base 256B).
// ---------------------------------------------------------------------------
__device__ __forceinline__ void preload_bc_slab(const float* __restrict__ xdbl,
                                                size_t m0, float* sBC) {
#pragma unroll
    for (int i0 = threadIdx.x * 4; i0 < CLEN * 32; i0 += 1024) {
        const int r  = i0 >> 5;
        const int cc = i0 & 31;
        async_g2l_b128(xdbl + (m0 + r) * XPN + DTRANK + cc, &sBC[i0]);
    }
    async_wait0();
    __syncthreads();
}

// ---------------------------------------------------------------------------
// Scan pass 1: per-chunk local scan (h0 = 0).  Emits h_local_end and the
// chunk propagator  P[n] = exp(A[n] * sum(dt))  (= prod of per-step dA).
// Grid: (DINNER/256, B*NCHUNK), block 256.
// ---------------------------------------------------------------------------
__global__ void __launch_bounds__(256)
scan_chunk_pass1(const float* __restrict__ delta, const float* __restrict__ u,
                 const float* __restrict__ xdbl, const float* __restrict__ A_log,
                 float* __restrict__ hend, float* __restrict__ prodA) {
    const int d  = blockIdx.x * 256 + threadIdx.x;
    const int bc = blockIdx.y;            // b*NCHUNK + c
    const int b  = bc >> 5;
    const int c  = bc & (NCHUNK - 1);
    const size_t m0 = (size_t)b * LSEQ + (size_t)c * CLEN;

    __shared__ __align__(16) float sBC[CLEN * 32];
    preload_bc_slab(xdbl, m0, sBC);

    float Ad[DSTATE], h[DSTATE];
#pragma unroll
    for (int n = 0; n < DSTATE; ++n) {
        Ad[n] = -expf(A_log[d * DSTATE + n]);
        h[n]  = 0.0f;
    }
    float dtsum = 0.0f;
    for (int l = 0; l < CLEN; ++l) {
        const float dt = delta[(m0 + l) * DINNER + d];
        const float uu = u[(m0 + l) * DINNER + d];
        dtsum += dt;
        const float du = dt * uu;
#pragma unroll
        for (int n = 0; n < DSTATE; ++n) {
            h[n] = expf(dt * Ad[n]) * h[n] + du * sBC[l * 32 + n];
        }
    }
    const size_t o = ((size_t)bc * DINNER + d) * DSTATE;
#pragma unroll
    for (int n = 0; n < DSTATE; ++n) {
        hend[o + n]  = h[n];
        prodA[o + n] = expf(Ad[n] * dtsum);
    }
}

// ---------------------------------------------------------------------------
// Scan pass 2: 32-step sequential combine per (b,d,n) -> per-chunk h_init.
// ---------------------------------------------------------------------------
__global__ void __launch_bounds__(256)
scan_combine(const float* __restrict__ hend, const float* __restrict__ prodA,
             float* __restrict__ hinit) {
    const int idx = blockIdx.x * 256 + threadIdx.x; // over B*DINNER*DSTATE
    const int n = idx & (DSTATE - 1);
    const int d = (idx >> 4) & (DINNER - 1);
    const int b = idx >> 15;                        // DINNER*DSTATE = 2^15
    float h = 0.0f;
    for (int c = 0; c < NCHUNK; ++c) {
        const size_t o = ((size_t)(b * NCHUNK + c) * DINNER + d) * DSTATE + n;
        hinit[o] = h;
        h = prodA[o] * h + hend[o];
    }
}

// ---------------------------------------------------------------------------
// Scan pass 3: re-scan each chunk from its true initial state; fuse
// y = (h.C) + u*D  and  y *= silu(z)  (z = xz[:, DINNER:2*DINNER]).
// ---------------------------------------------------------------------------
__global__ void __launch_bounds__(256)
scan_chunk_pass3(const float* __restrict__ delta, const float* __restrict__ u,
                 const float* __restrict__ xdbl, const float* __restrict__ xz,
                 const float* __restrict__ A_log, const float* __restrict__ Dv,
                 const float* __restrict__ hinit, float* __restrict__ ybuf) {
    const int d  = blockIdx.x * 256 + threadIdx.x;
    const int bc = blockIdx.y;
    const int b  = bc >> 5;
    const int c  = bc & (NCHUNK - 1);
    const size_t m0 = (size_t)b * LSEQ + (size_t)c * CLEN;

    __shared__ __align__(16) float sBC[CLEN * 32];
    preload_bc_slab(xdbl, m0, sBC);

    float Ad[DSTATE], h[DSTATE];
    const size_t o = ((size_t)bc * DINNER + d) * DSTATE;
#pragma unroll
    for (int n = 0; n < DSTATE; ++n) {
        Ad[n] = -expf(A_log[d * DSTATE + n]);
        h[n]  = hinit[o + n];
    }
    const float Dd = Dv[d];
    for (int l = 0; l < CLEN; ++l) {
        const size_t m = m0 + l;
        const float dt = delta[m * DINNER + d];
        const float uu = u[m * DINNER + d];
        const float du = dt * uu;
        float acc = 0.0f;
#pragma unroll
        for (int n = 0; n < DSTATE; ++n) {
            h[n] = expf(dt * Ad[n]) * h[n] + du * sBC[l * 32 + n];
            acc += h[n] * sBC[l * 32 + DSTATE + n];
        }
        const float zz = xz[m * (2 * DINNER) + DINNER + d];
        ybuf[m * DINNER + d] = (acc + uu * Dd) * silu_f(zz);
    }
}

// ---------------------------------------------------------------------------
// Launcher. Inputs: x, in_proj_w, conv_w, conv_b, x_proj_w, dt_proj_w,
// dt_proj_b, A_log, D, out_proj_w, proj_w, proj_b.
// ---------------------------------------------------------------------------
extern "C" void kernel_launch(void* const* d_in, const int* in_sizes, int n_in,
                              void* d_out, int out_size, void* d_ws,
                              size_t ws_size, hipStream_t stream) {
    const float* x         = (const float*)d_in[0];
    const float* in_proj_w = (const float*)d_in[1];
    const float* conv_w    = (const float*)d_in[2];
    const float* conv_b    = (const float*)d_in[3];
    const float* x_proj_w  = (const float*)d_in[4];
    const float* dt_proj_w = (const float*)d_in[5];
    const float* dt_proj_b = (const float*)d_in[6];
    const float* A_log     = (const float*)d_in[7];
    const float* Dvec      = (const float*)d_in[8];
    const float* out_proj_w= (const float*)d_in[9];
    const float* proj_w    = (const float*)d_in[10];
    const float* proj_b    = (const float*)d_in[11];
    float* out = (float*)d_out;
    float* ws  = (float*)d_ws;

    const size_t SUMSZ = (size_t)MB * NCHUNK * DINNER * DSTATE; // 4M floats
    size_t off = 0;
    float* xz    = ws + off; off += (size_t)MROWS * (2 * DINNER); // 128 MB
    float* ucv   = ws + off; off += (size_t)MROWS * DINNER;       //  64 MB
    float* xdbl  = ws + off; off += (size_t)MROWS * XPN;          //   3 MB
    float* delta = ws + off; off += (size_t)MROWS * DINNER;       //  64 MB
    float* ybuf  = ws + off; off += (size_t)MROWS * DINNER;       //  64 MB
    float* out1  = ws + off; off += (size_t)MROWS * DMODEL;       //  32 MB
    float* hend  = ws + off; off += SUMSZ;                        //  16 MB
    float* prodA = ws + off; off += SUMSZ;                        //  16 MB
    float* hinit = ws + off; off += SUMSZ;                        //  16 MB

    const dim3 blk(256);

    // 1) xz = xt @ in_proj_w^T  (transpose folded into A loads)
    wmma_gemm_f32<1, 0, 0><<<dim3((2 * DINNER) / 64, MROWS / 128), blk, 0,
                             stream>>>(x, in_proj_w, nullptr, xz, MROWS,
                                       2 * DINNER, DMODEL, 0);

    // 2) depthwise causal conv + SiLU
    {
        const size_t tot = (size_t)MROWS * DINNER;
        conv_silu_kernel<<<(unsigned)((tot + 255) / 256), blk, 0, stream>>>(
            xz, conv_w, conv_b, ucv);
    }

    // 3) x_dbl = u @ x_proj_w^T   (N = 96)
    wmma_gemm_f32<0, 0, 0><<<dim3((XPN + 63) / 64, MROWS / 128), blk, 0,
                             stream>>>(ucv, x_proj_w, nullptr, xdbl, MROWS,
                                       XPN, DINNER, DINNER);

    // 4) delta = softplus(dt_low @ dt_proj_w^T + dt_proj_b)
    wmma_gemm_f32<0, 1, 0><<<dim3(DINNER / 64, MROWS / 128), blk, 0, stream>>>(
        xdbl, dt_proj_w, dt_proj_b, delta, MROWS, DINNER, DTRANK, XPN);

    // 5) chunked selective scan
    scan_chunk_pass1<<<dim3(DINNER / 256, MB * NCHUNK), blk, 0, stream>>>(
        delta, ucv, xdbl, A_log, hend, prodA);
    scan_combine<<<dim3((unsigned)(SUMSZ / 256)), blk, 0, stream>>>(
        hend, prodA, hinit);
    scan_chunk_pass3<<<dim3(DINNER / 256, MB * NCHUNK), blk, 0, stream>>>(
        delta, ucv, xdbl, xz, A_log, Dvec, hinit, ybuf);

    // 6) out1 = y @ out_proj_w^T
    wmma_gemm_f32<0, 0, 0><<<dim3(DMODEL / 64, MROWS / 128), blk, 0, stream>>>(
        ybuf, out_proj_w, nullptr, out1, MROWS, DMODEL, DINNER, DINNER);

    // 7) out = (out1 @ proj_w^T + proj_b), stored transposed to (B, Dm, L)
    wmma_gemm_f32<0, 2, 1><<<dim3(DMODEL / 64, MROWS / 128), blk, 0, stream>>>(
        out1, proj_w, proj_b, out, MROWS, DMODEL, DMODEL, DMODEL);
}